// Loss_90400471646210
// MI455X (gfx1250) — compile-verified
//
#include <hip/hip_runtime.h>

typedef float v2f __attribute__((ext_vector_type(2)));
typedef float v8f __attribute__((ext_vector_type(8)));

#define N_SAMPLES        128
#define NUM_OBJECTS      8
#define NUM_POINTS       100000
#define GROUPS           (NUM_POINTS / 16)            // 6250 groups of 16 points
#define SLICES           16                           // blocks per sample
#define WAVES_PER_BLOCK  8
#define THREADS          (WAVES_PER_BLOCK * 32)       // 256 threads (8 wave32)
#define WAVES_PER_SAMPLE (SLICES * WAVES_PER_BLOCK)   // 128
#define GROUPS_PER_WAVE  ((GROUPS + WAVES_PER_SAMPLE - 1) / WAVES_PER_SAMPLE) // 49

// Kernel 1: per-(sample, slice) partial sums of the ADD/pm loss numerator.
// v_wmma_f32_16x16x4_f32: A = dR (3x3 zero-padded to 16x4),
// B = 16 points as columns (x,y,z,0). D rows 0..2 give moved x/y/z of each
// point in the SAME lane -> fused |.| + accumulate, no shuffles in the loop.
// Each wave streams a CONTIGUOUS chunk of point groups (+192B per iter) with a
// scalar (readfirstlane'd) trip count so EXEC stays full and loop control is SALU.
__global__ __launch_bounds__(THREADS)
void pm_partial_kernel(const int* __restrict__ obj_id,
                       const float* __restrict__ gtR,
                       const float* __restrict__ prR,
                       const float* __restrict__ pts,
                       float* __restrict__ ws)
{
    const int n     = blockIdx.x;      // sample
    const int slice = blockIdx.y;      // point-range slice
    const int tid   = threadIdx.x;
    const int wave  = tid >> 5;
    const int lane  = tid & 31;
    const int m     = lane & 15;
    const bool hi   = lane >= 16;

    // ---- Build A-matrix (16x4 f32): lane m<16 -> A[m][K=0],A[m][K=1];
    //      lane 16+m -> A[m][K=2],A[m][K=3]. Rows >=3 and col K=3 are zero. ----
    float a0 = 0.0f, a1 = 0.0f;
    if (m < 3) {
        const float* g = gtR + n * 9 + m * 3;
        const float* p = prR + n * 9 + m * 3;
        if (!hi) { a0 = p[0] - g[0]; a1 = p[1] - g[1]; }
        else     { a0 = p[2] - g[2]; /* K=3 column stays 0 */ }
    }
    v2f A; A[0] = a0; A[1] = a1;

    // Wave-uniform chunk of groups [g0, g0+iters), made explicitly scalar.
    const int wg = slice * WAVES_PER_BLOCK + wave;            // 0..127
    int g0    = wg * GROUPS_PER_WAVE;
    int rem   = GROUPS - g0;
    int iters = rem < GROUPS_PER_WAVE ? rem : GROUPS_PER_WAVE;
    if (iters < 0) iters = 0;
    g0    = __builtin_amdgcn_readfirstlane(g0);
    iters = __builtin_amdgcn_readfirstlane(iters);

    const int obj = obj_id[n];                                 // uniform
    const float* __restrict__ pb =
        pts + (size_t)obj * (size_t)(NUM_POINTS * 3) + (size_t)g0 * 48;

    // Per-lane element offsets inside one 16-point group (48 floats):
    // B VGPR0: lane n -> x of point n (K=0), lane 16+n -> z (K=2)
    // B VGPR1: lane n -> y of point n (K=1), lane 16+n -> 0  (K=3)
    const int off0 = 3 * m + (hi ? 2 : 0);
    const int off1 = 3 * m + 1;

    float acc = 0.0f;
    #pragma unroll 4
    for (int i = 0; i < iters; ++i) {
        float b0 = pb[off0];
        float y  = pb[off1];
        float b1 = hi ? 0.0f : y;
        v2f B; B[0] = b0; B[1] = b1;
        v8f c = {};  // C = 0 every iteration (abs must be applied per point)
        v8f d = __builtin_amdgcn_wmma_f32_16x16x4_f32(
                    /*neg_a=*/false, A, /*neg_b=*/false, B,
                    /*c_mod=*/(short)0, c, /*reuse_a=*/false, /*reuse_b=*/false);
        // lane n<16: d[0..2] = moved x,y,z of one point; lanes 16..31 -> 0
        acc += __builtin_fabsf(d[0]) + __builtin_fabsf(d[1]) + __builtin_fabsf(d[2]);
        pb += 48;  // next contiguous 16-point group (192 bytes)
    }

    // ---- wave32 reduction ----
    for (int o = 16; o > 0; o >>= 1) acc += __shfl_xor(acc, o, 32);

    __shared__ float wsum[WAVES_PER_BLOCK];
    if (lane == 0) wsum[wave] = acc;
    __syncthreads();
    if (tid == 0) {
        float s = 0.0f;
        #pragma unroll
        for (int w = 0; w < WAVES_PER_BLOCK; ++w) s += wsum[w];
        ws[n * SLICES + slice] = s;
    }
}

// Kernel 2: deterministic finalize. Sums the SLICES partials per sample,
// scales by 1/(P * diameter), and computes the two translation losses.
// Writes all 3*N_SAMPLES outputs (handles d_out poisoning).
__global__ __launch_bounds__(N_SAMPLES)
void finalize_kernel(const float* __restrict__ ws,
                     const int* __restrict__ obj_id,
                     const float* __restrict__ gt_t,
                     const float* __restrict__ pr_t,
                     const float* __restrict__ diam,
                     float* __restrict__ out)
{
    const int n = threadIdx.x;
    if (n >= N_SAMPLES) return;

    float s = 0.0f;
    #pragma unroll
    for (int i = 0; i < SLICES; ++i) s += ws[n * SLICES + i];
    out[n] = s / ((float)NUM_POINTS * diam[obj_id[n]]);

    float dx = gt_t[n * 3 + 0] - pr_t[n * 3 + 0];
    float dy = gt_t[n * 3 + 1] - pr_t[n * 3 + 1];
    float dz = gt_t[n * 3 + 2] - pr_t[n * 3 + 2];
    out[N_SAMPLES + n]     = __builtin_fabsf(dx) + __builtin_fabsf(dy);
    out[2 * N_SAMPLES + n] = __builtin_fabsf(dz);
}

extern "C" void kernel_launch(void* const* d_in, const int* in_sizes, int n_in,
                              void* d_out, int out_size, void* d_ws, size_t ws_size,
                              hipStream_t stream) {
    const int*   obj_id = (const int*)d_in[0];
    const float* gtR    = (const float*)d_in[1];
    const float* prR    = (const float*)d_in[2];
    const float* gt_t   = (const float*)d_in[3];
    const float* pr_t   = (const float*)d_in[4];
    const float* pts    = (const float*)d_in[5];
    const float* diam   = (const float*)d_in[6];
    float* out = (float*)d_out;
    float* ws  = (float*)d_ws;   // needs 128 * 16 floats = 8 KB

    dim3 grid(N_SAMPLES, SLICES);
    pm_partial_kernel<<<grid, THREADS, 0, stream>>>(obj_id, gtR, prR, pts, ws);
    finalize_kernel<<<1, N_SAMPLES, 0, stream>>>(ws, obj_id, gt_t, pr_t, diam, out);
}